// BEVScatter_52561809768796
// MI455X (gfx1250) — compile-verified
//
#include <hip/hip_runtime.h>

// BEV scatter (MI455X / gfx1250, wave32).
// Memory-bound: ~256MB output stores (NT, bypass L2) + ~51MB feature gathers
// (kept L2-resident) + 4MB index scratch. Ideal ~13.5us @ 23.3 TB/s.
// CDNA5 path: GLOBAL_LOAD_ASYNC_TO_LDS_B128 (ASYNCcnt) + LDS transpose.

#define BEV_H   512
#define BEV_W   512
#define HW      (BEV_H * BEV_W)      // 262144
#define NCH     64
#define TILE    64                    // cells per block in gather kernel
#define LDS_STRIDE 68                 // 64 + 4 pad: keeps 16B alignment (mod 4 == 0),
                                      // bank step 4 -> at worst 2-way conflict on reads

// ---------------------------------------------------------------------------
// Kernel 1: init last[] = -1 and a 64-float zero row (for invalid cells).
// ---------------------------------------------------------------------------
__global__ void bev_init_kernel(int* __restrict__ last, float* __restrict__ zrow,
                                int ncells) {
    int i = blockIdx.x * blockDim.x + threadIdx.x;
    if (i < ncells) __builtin_nontemporal_store(-1, &last[i]);
    if (i < NCH)    zrow[i] = 0.0f;
}

// ---------------------------------------------------------------------------
// Kernel 2: last-occurrence-wins dedup == atomicMax of pillar index per cell.
// coords are read exactly once -> non-temporal loads.
// ---------------------------------------------------------------------------
__global__ void bev_scatter_kernel(const int* __restrict__ coords,
                                   int* __restrict__ last, int P) {
    int p = blockIdx.x * blockDim.x + threadIdx.x;
    if (p >= P) return;
    int b = __builtin_nontemporal_load(&coords[3 * p + 0]);
    int r = __builtin_nontemporal_load(&coords[3 * p + 1]);
    int c = __builtin_nontemporal_load(&coords[3 * p + 2]);
    r = (r < 0) ? 0 : (r > BEV_H - 1 ? BEV_H - 1 : r);
    c = (c < 0) ? 0 : (c > BEV_W - 1 ? BEV_W - 1 : c);
    int key = b * HW + r * BEV_W + c;
    atomicMax(&last[key], p);          // global_atomic_max_i32, no return needed
}

// ---------------------------------------------------------------------------
// Kernel 3: per 64-cell tile: async-copy 64 feature rows (256B each, fully
// contiguous) into LDS, then write the transpose out with coalesced NT stores.
// One block = 256 threads = 8 waves.
// ---------------------------------------------------------------------------
__global__ void bev_gather_kernel(const float* __restrict__ feats,
                                  const int*   __restrict__ last,
                                  const float* __restrict__ zrow,
                                  float*       __restrict__ out) {
    __shared__ float tile[TILE * LDS_STRIDE];   // 64 rows x 68 dwords = 4352 floats (17408 B)

    const int blk = blockIdx.x;           // 0 .. B*(HW/64)-1
    const int b   = blk >> 12;            // HW/TILE = 4096 tiles per batch
    const int hw0 = (blk & 4095) << 6;    // first cell of this tile
    const int t   = threadIdx.x;

    const int* lastB = last + b * HW + hw0;

    // ---- Phase 1: 64 cells x 16 chunks of 16B; 1024 async b128 copies -----
    // idx -> (cell = idx>>4, chunk = idx&15). Each 16-lane group streams one
    // contiguous 256B feature row straight into LDS (no VGPR staging).
    // Default RT policy: feature rows are re-read across tiles -> keep in L2.
#pragma unroll
    for (int i = 0; i < 4; ++i) {
        int idx   = i * 256 + t;                 // 0..1023
        int cell  = idx >> 4;
        int chunk = idx & 15;
        int l     = lastB[cell];
        const float* src = (l >= 0) ? (feats + (size_t)l * NCH + chunk * 4)
                                    : (zrow + chunk * 4);
        unsigned long long gaddr = (unsigned long long)(const void*)src;
        unsigned ldsoff = (unsigned)(size_t)(&tile[cell * LDS_STRIDE + chunk * 4]);
        asm volatile("global_load_async_to_lds_b128 %0, %1, off"
                     :: "v"(ldsoff), "v"(gaddr) : "memory");
    }
    asm volatile("s_wait_asynccnt 0x0" ::: "memory");
    __syncthreads();

    // ---- Phase 2: transposed, fully coalesced store stream ----------------
    // lane -> cell (contiguous hw), so each wave writes a contiguous 128B run.
    // Output (256MB) exceeds L2 (192MB): store non-temporal so the write
    // stream does not evict the L2-resident feature table.
    const int cell = t & 63;
    const int chb  = t >> 6;              // 0..3
    float* outB = out + (size_t)b * NCH * HW + hw0;
#pragma unroll
    for (int i = 0; i < 16; ++i) {
        int ch = chb + 4 * i;
        __builtin_nontemporal_store(tile[cell * LDS_STRIDE + ch],
                                    &outB[(size_t)ch * HW + cell]);
    }
}

// ---------------------------------------------------------------------------
extern "C" void kernel_launch(void* const* d_in, const int* in_sizes, int n_in,
                              void* d_out, int out_size, void* d_ws, size_t ws_size,
                              hipStream_t stream) {
    const float* feats  = (const float*)d_in[0];   // (P, 64) fp32
    const int*   coords = (const int*)d_in[1];     // (P, 3)  int32
    float*       out    = (float*)d_out;           // (B, 64, 512, 512) fp32

    const int P = in_sizes[0] / NCH;
    const int B = out_size / (NCH * HW);           // derive batch on host
    const int ncells = B * HW;

    // d_ws layout: [ last : ncells * int ][ zero row : 64 floats ]
    int*   last = (int*)d_ws;
    float* zrow = (float*)((char*)d_ws + (size_t)ncells * sizeof(int));

    bev_init_kernel<<<(ncells + 255) / 256, 256, 0, stream>>>(last, zrow, ncells);
    bev_scatter_kernel<<<(P + 255) / 256, 256, 0, stream>>>(coords, last, P);

    const int ntiles = ncells / TILE;              // one block per 64-cell tile
    bev_gather_kernel<<<ntiles, 256, 0, stream>>>(feats, last, zrow, out);
}